// MahalanobisMask_34969623724506
// MI455X (gfx1250) — compile-verified
//
#include <hip/hip_runtime.h>
#include <math.h>

// Problem constants (from reference): B=16, C=128, L=512, F=L/2+1=257
#define BB   16
#define CC   128
#define LL   512
#define FF   257
#define FP   272           // F padded to a multiple of 16 (zero padding -> exact results)
#define MR   (BB*CC)       // 2048 rows of xf / Y

typedef float v2f __attribute__((ext_vector_type(2)));
typedef float v8f __attribute__((ext_vector_type(8)));

// ---------------- WMMA f32 16x16x4 fragment helpers (ISA 7.12.2 layouts) ----
// A 16x4 f32: lanes 0-15 -> rows M=0..15 holding {K0,K1}; lanes 16-31 -> {K2,K3}
__device__ __forceinline__ v2f ldA(const float* M, int ld, int m0, int k0, int lane) {
    int row = m0 + (lane & 15);
    int kk  = k0 + ((lane >> 4) << 1);
    v2f r; r.x = M[row * ld + kk]; r.y = M[row * ld + kk + 1]; return r;
}
// B 4x16 f32: VGPR0 = rows K0 (lanes 0-15) / K2 (lanes 16-31); VGPR1 = K1 / K3
__device__ __forceinline__ v2f ldB(const float* M, int ld, int k0, int n0, int lane) {
    int col = n0 + (lane & 15);
    int kk  = k0 + ((lane >> 4) << 1);
    v2f r; r.x = M[kk * ld + col]; r.y = M[(kk + 1) * ld + col]; return r;
}
// B fragment where logical B[k][n] = M[rowbase+n][k]  (transposed source, for Gram)
__device__ __forceinline__ v2f ldBT(const float* M, int ld, int k0, int rowbase, int n0, int lane) {
    int col = n0 + (lane & 15);
    int kk  = k0 + ((lane >> 4) << 1);
    v2f r; r.x = M[(rowbase + col) * ld + kk]; r.y = M[(rowbase + col) * ld + kk + 1]; return r;
}
__device__ __forceinline__ v8f wmma4(v2f a, v2f b, v8f c) {
    // 8 args: (neg_a, A, neg_b, B, c_mod, C, reuse_a, reuse_b)
    return __builtin_amdgcn_wmma_f32_16x16x4_f32(false, a, false, b, (short)0, c, false, false);
}

// ---------------- K0a: DFT basis (cos / sin), zero-padded columns ------------
__global__ void k_basis(float* __restrict__ bc, float* __restrict__ bs) {
    int idx = blockIdx.x * blockDim.x + threadIdx.x;
    if (idx >= LL * FP) return;
    int n = idx / FP, f = idx % FP;
    float c = 0.f, s = 0.f;
    if (f < FF) {
        int ph = (n * f) & (LL - 1);                       // (n*f) mod 512, exact
        float ang = 6.28318530717958647692f * (float)ph * (1.0f / (float)LL);
        __sincosf(ang, &s, &c);                            // arg in [0, 2*pi): accurate
    }
    bc[idx] = c; bs[idx] = s;
}

// ---------------- K0b: pack A (FFxFF row-major, A[g][f]) -> At[f][g] padded --
__global__ void k_packA(const float* __restrict__ A, float* __restrict__ At) {
    int idx = blockIdx.x * blockDim.x + threadIdx.x;
    if (idx >= FP * FP) return;
    int f = idx / FP, g = idx % FP;
    At[idx] = (f < FF && g < FF) ? A[g * FF + f] : 0.f;
}

// ---------------- K1: xfabs = |DFT(x)| via two WMMA GEMMs --------------------
// M=2048, N=FP, K=512. One 16x16 tile per wave, 8 waves per block.
__global__ void k_dft(const float* __restrict__ x, const float* __restrict__ bc,
                      const float* __restrict__ bs, float* __restrict__ xf) {
    int lane = threadIdx.x & 31;
    int tile = blockIdx.x * 8 + (threadIdx.x >> 5);
    int m0 = (tile / (FP / 16)) * 16;
    int n0 = (tile % (FP / 16)) * 16;
    v8f aR = {}; v8f aI = {};
    for (int k0 = 0; k0 < LL; k0 += 4) {
        v2f a  = ldA(x,  LL, m0, k0, lane);
        v2f br = ldB(bc, FP, k0, n0, lane);
        v2f bi = ldB(bs, FP, k0, n0, lane);
        aR = wmma4(a, br, aR);
        aI = wmma4(a, bi, aI);
    }
    int col = n0 + (lane & 15);
#pragma unroll
    for (int r = 0; r < 8; ++r) {
        int row = m0 + r + ((lane >> 4) << 3);
        float re = aR[r], im = aI[r];
        xf[row * FP + col] = sqrtf(re * re + im * im);     // padded cols yield 0
    }
}

// ---------------- K2: Y = xfabs @ At  (M=2048, N=FP, K=FP) -------------------
__global__ void k_proj(const float* __restrict__ xf, const float* __restrict__ At,
                       float* __restrict__ Y) {
    int lane = threadIdx.x & 31;
    int tile = blockIdx.x * 8 + (threadIdx.x >> 5);
    int m0 = (tile / (FP / 16)) * 16;
    int n0 = (tile % (FP / 16)) * 16;
    v8f acc = {};
    for (int k0 = 0; k0 < FP; k0 += 4) {
        v2f a = ldA(xf, FP, m0, k0, lane);
        v2f b = ldB(At, FP, k0, n0, lane);
        acc = wmma4(a, b, acc);
    }
    int col = n0 + (lane & 15);
#pragma unroll
    for (int r = 0; r < 8; ++r) {
        int row = m0 + r + ((lane >> 4) << 3);
        Y[row * FP + col] = acc[r];
    }
}

// ---------------- K3: per-batch Gram G[b] = Yb @ Yb^T (128x128, K=FP) --------
__global__ void k_gram(const float* __restrict__ Y, float* __restrict__ G) {
    int lane = threadIdx.x & 31;
    int tile = blockIdx.x * 8 + (threadIdx.x >> 5);     // 0..1023 = 16 batches * 64 tiles
    int b  = tile >> 6;
    int t  = tile & 63;
    int m0 = (t >> 3) << 4;
    int n0 = (t & 7) << 4;
    int rb = b * CC;
    v8f acc = {};
    for (int k0 = 0; k0 < FP; k0 += 4) {
        v2f a  = ldA (Y, FP, rb + m0, k0, lane);
        v2f bt = ldBT(Y, FP, k0, rb, n0, lane);         // B[k][n] = Y[rb+n][k]
        acc = wmma4(a, bt, acc);
    }
    int col = n0 + (lane & 15);
#pragma unroll
    for (int r = 0; r < 8; ++r) {
        int row = m0 + r + ((lane >> 4) << 3);
        G[b * CC * CC + row * CC + col] = acc[r];
    }
}

// ---------------- K4: fused elementwise + row-max + Gumbel hard mask ---------
// dist[b,i,j] = G_ii + G_jj - 2*G_ij ; one block per (b,i) row, 128 threads.
__global__ void k_mask(const float* __restrict__ G, const float* __restrict__ gu,
                       float* __restrict__ out) {
    __shared__ float red[CC];
    int bi = blockIdx.x;               // b*CC + i
    int b  = bi >> 7, i = bi & 127;
    int j  = threadIdx.x;
    const float* Gb = G + b * CC * CC;
    float gii = Gb[i * CC + i];
    float gjj = Gb[j * CC + j];
    float gij = Gb[i * CC + j];
    float dist = fmaxf(gii + gjj - 2.f * gij, 0.f);
    float e = (j == i) ? 0.f : 1.f / (dist + 3e-8f);
    red[j] = e;
    __syncthreads();
#pragma unroll
    for (int s = 64; s > 0; s >>= 1) {
        if (j < s) red[j] = fmaxf(red[j], red[j + s]);
        __syncthreads();
    }
    float mx = red[0];
    float p = e / (mx + 1e-8f);
    if (j == i) p = 1.f;               // exp_dist diag zeroed, then +eye
    p *= 0.99f;
    float logits = logf(p + 1e-8f) - logf(1.f - p + 1e-8f);
    const float* u = gu + ((size_t)bi * CC + j) * 2;
    float g0 = -logf(-logf(u[0] + 1e-20f) + 1e-20f);
    float g1 = -logf(-logf(u[1] + 1e-20f) + 1e-20f);
    // samples = one_hot(argmax(softmax(...))) -> hard compare; ties -> index 0
    out[(size_t)bi * CC + j] = (logits + g0 >= -logits + g1) ? 1.f : 0.f;
}

// ---------------- launcher ---------------------------------------------------
extern "C" void kernel_launch(void* const* d_in, const int* in_sizes, int n_in,
                              void* d_out, int out_size, void* d_ws, size_t ws_size,
                              hipStream_t stream) {
    (void)in_sizes; (void)n_in; (void)out_size; (void)ws_size;
    const float* x  = (const float*)d_in[0];   // (16,128,512)
    const float* A  = (const float*)d_in[1];   // (257,257)
    const float* gu = (const float*)d_in[2];   // (16,128,128,2)

    float* ws = (float*)d_ws;                  // ~6.9 MB of f32 scratch
    float* bc = ws;                            // 512*272
    float* bs = bc + LL * FP;                  // 512*272
    float* xf = bs + LL * FP;                  // 2048*272
    float* At = xf + MR * FP;                  // 272*272
    float* Y  = At + FP * FP;                  // 2048*272
    float* G  = Y  + MR * FP;                  // 16*128*128

    k_basis<<<(LL * FP + 255) / 256, 256, 0, stream>>>(bc, bs);
    k_packA<<<(FP * FP + 255) / 256, 256, 0, stream>>>(A, At);
    // (2048/16)*(272/16) = 2176 tiles, 8 waves/block -> 272 blocks
    k_dft <<<272, 256, 0, stream>>>(x, bc, bs, xf);
    k_proj<<<272, 256, 0, stream>>>(xf, At, Y);
    // 16 batches * 64 tiles = 1024 tiles -> 128 blocks
    k_gram<<<128, 256, 0, stream>>>(Y, G);
    k_mask<<<MR, CC, 0, stream>>>(G, gu, (float*)d_out);
}